// DSEA_30975304138875
// MI455X (gfx1250) — compile-verified
//
#include <hip/hip_runtime.h>
#include <math.h>

#define N_NODES 100000
#define N_REL   1000
#define N_E     400000
#define N_EALL  500000
#define EH      300
#define RH      100

typedef float v2f __attribute__((ext_vector_type(2)));
typedef float v8f __attribute__((ext_vector_type(8)));

__device__ __forceinline__ float leakyf(float x) { return x > 0.f ? x : 0.01f * x; }

// float atomic max via signed/unsigned integer ordering trick (init to -inf)
__device__ __forceinline__ void atomicMaxFloat(float* addr, float v) {
    if (v >= 0.f) atomicMax((int*)addr, __float_as_int(v));
    else          atomicMin((unsigned int*)addr, __float_as_uint(v));
}

// ---------------------------------------------------------------- utilities
__global__ __launch_bounds__(256)
void fill_f32(float* __restrict__ p, float v, long long n) {
    long long i = (long long)blockIdx.x * blockDim.x + threadIdx.x;
    if (i < n) p[i] = v;
}

// Wp[k, 0:ncp] = W[k, 0:ncols] zero-padded to ncp columns
__global__ __launch_bounds__(256)
void pad_w(const float* __restrict__ W, int K, int ncols,
           float* __restrict__ Wp, int ncp) {
    int idx = blockIdx.x * blockDim.x + threadIdx.x;
    int k = idx / ncp, c = idx % ncp;
    if (k >= K) return;
    Wp[idx] = (c < ncols) ? W[k * ncols + c] : 0.f;
}

__global__ __launch_bounds__(256)
void deg_kernel(const int* __restrict__ dst, int E, float* __restrict__ deg) {
    int e = blockIdx.x * blockDim.x + threadIdx.x;
    if (e < E) atomicAdd(&deg[dst[e]], 1.0f);
}

__global__ __launch_bounds__(256)
void rsqrt_kernel(const float* __restrict__ deg, float* __restrict__ dinv, int n) {
    int i = blockIdx.x * blockDim.x + threadIdx.x;
    if (i < n) dinv[i] = rsqrtf(deg[i]);   // deg==0 -> +inf, matches pow(deg,-0.5)
}

// ---------------------------------------------------------------- GCN scatter
// agg[i,:] += dinv[j]*dinv[i] * X[j,:]   (float4 vectorized, EH=300 -> 75 quads)
__global__ __launch_bounds__(256)
void gcn_scatter(const float* __restrict__ X,
                 const int* __restrict__ src, const int* __restrict__ dst,
                 const float* __restrict__ dinv,
                 float* __restrict__ agg, int E) {
    const int Q = EH / 4;  // 75
    long long idx = (long long)blockIdx.x * blockDim.x + threadIdx.x;
    int q = (int)(idx % Q);
    long long e = idx / Q;
    if (e >= E) return;
    int j = src[e], i = dst[e];
    float norm = dinv[j] * dinv[i];
    float4 v = ((const float4*)(X + (size_t)j * EH))[q];
    float* o = agg + (size_t)i * EH + (size_t)q * 4;
    atomicAdd(o + 0, norm * v.x);
    atomicAdd(o + 1, norm * v.y);
    atomicAdd(o + 2, norm * v.z);
    atomicAdd(o + 3, norm * v.w);
}

// ---------------------------------------------------------------- WMMA GEMM
// D = X[M x K] @ W[K x ncols], fp32 WMMA 16x16x4.
// W is pre-padded to NCP columns (NCP % (16*NT) == 0), so the hot loop has no
// masking and only compile-time immediate offsets off two streaming pointers.
// Each wave computes a 16 x (16*NT) strip: one A fragment feeds NT WMMAs/step.
// mode 0: OUT = D
// mode 1: highway epilogue: gate = leaky(D + 1e-8);
//         OUT = leaky(gate * relu(AGG) + (1-gate) * X)   (ncols == K here)
// Requirements (met here): M % 16 == 0, K % 4 == 0.
template <int NT, int NCP>
__global__ __launch_bounds__(128)
void gemm_wmma(const float* __restrict__ X, int ldx,
               const float* __restrict__ Wp,          // K x NCP, zero-padded
               const float* __restrict__ AGG, int ldagg,
               float* __restrict__ OUT, int ldo,
               int M, int K, int ncols, int mode) {
    constexpr int groupsN = NCP / (16 * NT);
    int wave = blockIdx.x * (blockDim.x >> 5) + (threadIdx.x >> 5);
    int lane = threadIdx.x & 31;
    int tm = wave / groupsN;
    int gn = wave % groupsN;
    if (tm * 16 >= M) return;

    int half = lane >> 4;          // 0: lanes 0-15, 1: lanes 16-31
    int r    = lane & 15;
    int row  = tm * 16 + r;        // A-matrix row held by this lane
    int cn0  = gn * 16 * NT + r;   // first D column held by this lane

    const float* ap = X + (size_t)row * ldx + 2 * half;          // A stream
    const float* bp = Wp + (size_t)(2 * half) * NCP + cn0;       // B stream

    v8f c[NT];
    #pragma unroll
    for (int u = 0; u < NT; ++u) c[u] = (v8f){};

    for (int k = 0; k < K; k += 4) {
        v2f a;                     // A: K = 2*(lane/16) + vgpr
        a.x = ap[0];
        a.y = ap[1];
        #pragma unroll
        for (int u = 0; u < NT; ++u) {
            v2f b;                 // B: same K split, immediate offsets only
            b.x = bp[u * 16];
            b.y = bp[u * 16 + NCP];
            c[u] = __builtin_amdgcn_wmma_f32_16x16x4_f32(false, a, false, b,
                                                         (short)0, c[u],
                                                         false, false);
        }
        ap += 4;
        bp += (size_t)4 * NCP;
    }

    #pragma unroll
    for (int u = 0; u < NT; ++u) {
        int cn = cn0 + u * 16;
        if (cn >= ncols) continue;
        #pragma unroll
        for (int v = 0; v < 8; ++v) {
            int m = tm * 16 + v + 8 * half;   // C/D: M = vgpr + 8*(lane/16)
            float d = c[u][v];
            if (mode == 1) {
                float gate = leakyf(d + 1e-8f);
                float x1 = X[(size_t)m * ldx + cn];
                float ag = AGG[(size_t)m * ldagg + cn];
                ag = ag > 0.f ? ag : 0.f;              // relu (gcn output)
                OUT[(size_t)m * ldo + cn] =
                    leakyf(gate * ag + (1.f - gate) * x1);
            } else {
                OUT[(size_t)m * ldo + cn] = d;
            }
        }
    }
}

// ---------------------------------------------------------------- row . vec
// out[n] = dot(X[n, 0:K], a)    one wave (32 lanes) per row
__global__ __launch_bounds__(256)
void rowdot(const float* __restrict__ X, int ldx, const float* __restrict__ a,
            int K, float* __restrict__ out, int nrows) {
    int wave = blockIdx.x * (blockDim.x >> 5) + (threadIdx.x >> 5);
    int lane = threadIdx.x & 31;
    if (wave >= nrows) return;
    const float* xr = X + (size_t)wave * ldx;
    float s = 0.f;
    for (int k = lane; k < K; k += 32) s += xr[k] * a[k];
    #pragma unroll
    for (int off = 16; off; off >>= 1) s += __shfl_xor(s, off, 32);
    if (lane == 0) out[wave] = s;
}

// ---------------------------------------------------------------- seg softmax
// pass 1: e[t] = leaky(sA[iA[t]] + sB[iB[t]]); m[seg] = max
__global__ __launch_bounds__(256)
void edge_score(const float* __restrict__ sA, const int* __restrict__ iA,
                const float* __restrict__ sB, const int* __restrict__ iB,
                const int* __restrict__ seg, int E,
                float* __restrict__ e, float* __restrict__ m) {
    int t = blockIdx.x * blockDim.x + threadIdx.x;
    if (t >= E) return;
    float v = leakyf(sA[iA[t]] + sB[iB[t]]);
    e[t] = v;
    atomicMaxFloat(&m[seg[t]], v);
}

// pass 2: e[t] = exp(e[t]-m[seg]); s[seg] += e[t]
__global__ __launch_bounds__(256)
void edge_expsum(const int* __restrict__ seg, int E, float* __restrict__ e,
                 const float* __restrict__ m, float* __restrict__ s) {
    int t = blockIdx.x * blockDim.x + threadIdx.x;
    if (t >= E) return;
    float ex = expf(e[t] - m[seg[t]]);
    e[t] = ex;
    atomicAdd(&s[seg[t]], ex);
}

// pass 3: OUT[seg[t], :] += (e[t]/s[seg[t]]) * SRC[g[t], :]   (float4, width4 quads)
__global__ __launch_bounds__(256)
void edge_scatter4(const float* __restrict__ e, const float* __restrict__ s,
                   const int* __restrict__ seg, const int* __restrict__ g,
                   const float* __restrict__ SRC, int lds,
                   float* __restrict__ OUT, int ldo, int width4, int E) {
    long long idx = (long long)blockIdx.x * blockDim.x + threadIdx.x;
    int q = (int)(idx % width4);
    long long t = idx / width4;
    if (t >= E) return;
    int sg = seg[t];
    float alpha = e[t] / s[sg];
    float4 v = ((const float4*)(SRC + (size_t)g[t] * lds))[q];
    float* o = OUT + (size_t)sg * ldo + (size_t)q * 4;
    atomicAdd(o + 0, alpha * v.x);
    atomicAdd(o + 1, alpha * v.y);
    atomicAdd(o + 2, alpha * v.z);
    atomicAdd(o + 3, alpha * v.w);
}

// elementwise leaky over a column slab of a row-major matrix
__global__ __launch_bounds__(256)
void leaky_slab(float* __restrict__ out, long long nrows, int rowstride,
                int coloff, int width) {
    long long idx = (long long)blockIdx.x * blockDim.x + threadIdx.x;
    int c = (int)(idx % width);
    long long r = idx / width;
    if (r >= nrows) return;
    float* p = out + r * rowstride + coloff + c;
    *p = leakyf(*p);
}

// ---------------------------------------------------------------- launcher
static inline int cdiv(long long a, long long b) { return (int)((a + b - 1) / b); }

extern "C" void kernel_launch(void* const* d_in, const int* in_sizes, int n_in,
                              void* d_out, int out_size, void* d_ws, size_t ws_size,
                              hipStream_t stream) {
    const float* x_e   = (const float*)d_in[0];
    const int*   ei    = (const int*)d_in[1];   // [2, E]
    const int*   rel   = (const int*)d_in[2];
    const int*   eia   = (const int*)d_in[3];   // [2, E_ALL]
    /* d_in[4] rel_all: unused by reference */
    const float* hw1_w = (const float*)d_in[5];
    const float* hw2_w = (const float*)d_in[6];
    const float* w_h   = (const float*)d_in[7];
    const float* w_t   = (const float*)d_in[8];
    const float* a_h1  = (const float*)d_in[9];
    const float* a_h2  = (const float*)d_in[10];
    const float* a_t1  = (const float*)d_in[11];
    const float* a_t2  = (const float*)d_in[12];
    const float* gr_ah = (const float*)d_in[13];
    const float* gr_ar = (const float*)d_in[14];
    const float* g_ai  = (const float*)d_in[15];
    const float* g_aj  = (const float*)d_in[16];

    const int* h     = ei;                 // edge_index[0]
    const int* t     = ei + N_E;           // edge_index[1]
    const int* j_all = eia;                // edge_index_all[0]
    const int* i_all = eia + N_EALL;       // edge_index_all[1]

    float* out = (float*)d_out;            // N x 1200 row-major

    constexpr int NCP_EH = 320;            // 300 cols padded (5 groups of 64)
    constexpr int NCP_RH = 128;            // 100 cols padded (2 groups of 64)

    // ---- workspace carve-up (floats) ----
    float* ws = (float*)d_ws;
    size_t o = 0;
    float* agg   = ws + o; o += (size_t)N_NODES * EH;
    float* xtmp  = ws + o; o += (size_t)N_NODES * EH;
    float* xrh   = ws + o; o += (size_t)N_NODES * RH;
    float* xrt   = ws + o; o += (size_t)N_NODES * RH;
    float* deg   = ws + o; o += N_NODES;
    float* dinv  = ws + o; o += N_NODES;
    float* sh1   = ws + o; o += N_NODES;
    float* sh2   = ws + o; o += N_NODES;
    float* st1   = ws + o; o += N_NODES;
    float* st2   = ws + o; o += N_NODES;
    float* xesc  = ws + o; o += N_NODES;   // x @ gr_ah
    float* gi    = ws + o; o += N_NODES;
    float* gj    = ws + o; o += N_NODES;
    float* mg    = ws + o; o += N_NODES;
    float* sg    = ws + o; o += N_NODES;
    float* e1    = ws + o; o += N_E;
    float* e2    = ws + o; o += N_E;
    float* er    = ws + o; o += N_E;
    float* eg    = ws + o; o += N_EALL;
    float* x_r   = ws + o; o += (size_t)N_REL * RH;
    float* xrsc  = ws + o; o += N_REL;     // x_r @ gr_ar
    float* m1    = ws + o; o += N_REL;     // m1,m2,mr contiguous
    float* m2    = ws + o; o += N_REL;
    float* mr    = ws + o; o += N_REL;
    float* s1    = ws + o; o += N_REL;     // s1,s2,sr contiguous
    float* s2    = ws + o; o += N_REL;
    float* sr    = ws + o; o += N_REL;
    float* wp1   = ws + o; o += (size_t)EH * NCP_EH;   // padded hw1_w
    float* wp2   = ws + o; o += (size_t)EH * NCP_EH;   // padded hw2_w
    float* wph   = ws + o; o += (size_t)EH * NCP_RH;   // padded w_h
    float* wpt   = ws + o; o += (size_t)EH * NCP_RH;   // padded w_t
    (void)s2; (void)m2; (void)ws_size; (void)n_in; (void)in_sizes;

    const float NEG_INF = -__builtin_inff();

    // ---- init ----
    hipMemsetAsync(d_out, 0, (size_t)out_size * sizeof(float), stream);
    hipMemsetAsync(agg, 0, (size_t)N_NODES * EH * sizeof(float), stream);
    hipMemsetAsync(deg, 0, (size_t)N_NODES * sizeof(float), stream);
    hipMemsetAsync(x_r, 0, (size_t)N_REL * RH * sizeof(float), stream);
    hipMemsetAsync(s1, 0, (size_t)3 * N_REL * sizeof(float), stream);
    hipMemsetAsync(sg, 0, (size_t)N_NODES * sizeof(float), stream);
    fill_f32<<<cdiv(3 * N_REL, 256), 256, 0, stream>>>(m1, NEG_INF, 3 * N_REL);
    fill_f32<<<cdiv(N_NODES, 256), 256, 0, stream>>>(mg, NEG_INF, N_NODES);

    // ---- pad weights once (tiny) ----
    pad_w<<<cdiv((long long)EH * NCP_EH, 256), 256, 0, stream>>>(hw1_w, EH, EH, wp1, NCP_EH);
    pad_w<<<cdiv((long long)EH * NCP_EH, 256), 256, 0, stream>>>(hw2_w, EH, EH, wp2, NCP_EH);
    pad_w<<<cdiv((long long)EH * NCP_RH, 256), 256, 0, stream>>>(w_h, EH, RH, wph, NCP_RH);
    pad_w<<<cdiv((long long)EH * NCP_RH, 256), 256, 0, stream>>>(w_t, EH, RH, wpt, NCP_RH);

    // ---- degree / norm ----
    deg_kernel<<<cdiv(N_EALL, 256), 256, 0, stream>>>(i_all, N_EALL, deg);
    rsqrt_kernel<<<cdiv(N_NODES, 256), 256, 0, stream>>>(deg, dinv, N_NODES);

    // ---- gcn #1 + highway #1 -> xtmp ----
    gcn_scatter<<<cdiv((long long)N_EALL * (EH / 4), 256), 256, 0, stream>>>(
        x_e, j_all, i_all, dinv, agg, N_EALL);
    {
        int waves = (N_NODES / 16) * (NCP_EH / 64);        // 5 groups
        gemm_wmma<4, NCP_EH><<<cdiv(waves, 4), 128, 0, stream>>>(
            x_e, EH, wp1, agg, EH, xtmp, EH, N_NODES, EH, EH, 1);
    }

    // ---- gcn #2 + highway #2 -> out[:, 0:300] ----
    hipMemsetAsync(agg, 0, (size_t)N_NODES * EH * sizeof(float), stream);
    gcn_scatter<<<cdiv((long long)N_EALL * (EH / 4), 256), 256, 0, stream>>>(
        xtmp, j_all, i_all, dinv, agg, N_EALL);
    {
        int waves = (N_NODES / 16) * (NCP_EH / 64);
        gemm_wmma<4, NCP_EH><<<cdiv(waves, 4), 128, 0, stream>>>(
            xtmp, EH, wp2, agg, EH, out, 4 * EH, N_NODES, EH, EH, 1);
    }

    // ---- gat_e: x_r_h = x @ w_h, x_r_t = x @ w_t ----
    {
        int waves = (N_NODES / 16) * (NCP_RH / 64);        // 2 groups
        gemm_wmma<4, NCP_RH><<<cdiv(waves, 4), 128, 0, stream>>>(
            out, 4 * EH, wph, nullptr, 0, xrh, RH, N_NODES, EH, RH, 0);
        gemm_wmma<4, NCP_RH><<<cdiv(waves, 4), 128, 0, stream>>>(
            out, 4 * EH, wpt, nullptr, 0, xrt, RH, N_NODES, EH, RH, 0);
    }
    {
        int rb = cdiv((long long)N_NODES * 32, 256);
        rowdot<<<rb, 256, 0, stream>>>(xrh, RH, a_h1, RH, sh1, N_NODES);
        rowdot<<<rb, 256, 0, stream>>>(xrt, RH, a_h2, RH, sh2, N_NODES);
        rowdot<<<rb, 256, 0, stream>>>(xrh, RH, a_t1, RH, st1, N_NODES);
        rowdot<<<rb, 256, 0, stream>>>(xrt, RH, a_t2, RH, st2, N_NODES);
    }
    {
        int eb = cdiv(N_E, 256);
        edge_score<<<eb, 256, 0, stream>>>(sh1, h, sh2, t, rel, N_E, e1, m1);
        edge_score<<<eb, 256, 0, stream>>>(st1, h, st2, t, rel, N_E, e2, m2);
        edge_expsum<<<eb, 256, 0, stream>>>(rel, N_E, e1, m1, s1);
        edge_expsum<<<eb, 256, 0, stream>>>(rel, N_E, e2, m2, s2);
        int sb = cdiv((long long)N_E * (RH / 4), 256);
        edge_scatter4<<<sb, 256, 0, stream>>>(e1, s1, rel, h, xrh, RH,
                                              x_r, RH, RH / 4, N_E);
        edge_scatter4<<<sb, 256, 0, stream>>>(e2, s2, rel, t, xrt, RH,
                                              x_r, RH, RH / 4, N_E);
    }

    // ---- gat_r -> out[:, 300:600] (rows >= R stay zero) ----
    rowdot<<<cdiv((long long)N_NODES * 32, 256), 256, 0, stream>>>(
        out, 4 * EH, gr_ah, EH, xesc, N_NODES);
    rowdot<<<cdiv((long long)N_REL * 32, 256), 256, 0, stream>>>(
        x_r, RH, gr_ar, RH, xrsc, N_REL);
    {
        int eb = cdiv(N_E, 256);
        edge_score<<<eb, 256, 0, stream>>>(xesc, h, xrsc, rel, rel, N_E, er, mr);
        edge_expsum<<<eb, 256, 0, stream>>>(rel, N_E, er, mr, sr);
        int sb = cdiv((long long)N_E * (EH / 4), 256);
        edge_scatter4<<<sb, 256, 0, stream>>>(er, sr, rel, t, out, 4 * EH,
                                              out + EH, 4 * EH, EH / 4, N_E);
    }

    // ---- gat over out[:, 0:600] -> out[:, 600:1200], then leaky ----
    rowdot<<<cdiv((long long)N_NODES * 32, 256), 256, 0, stream>>>(
        out, 4 * EH, g_ai, 2 * EH, gi, N_NODES);
    rowdot<<<cdiv((long long)N_NODES * 32, 256), 256, 0, stream>>>(
        out, 4 * EH, g_aj, 2 * EH, gj, N_NODES);
    {
        int eb = cdiv(N_EALL, 256);
        edge_score<<<eb, 256, 0, stream>>>(gi, i_all, gj, j_all, i_all,
                                           N_EALL, eg, mg);
        edge_expsum<<<eb, 256, 0, stream>>>(i_all, N_EALL, eg, mg, sg);
        int sb = cdiv((long long)N_EALL * (2 * EH / 4), 256);
        edge_scatter4<<<sb, 256, 0, stream>>>(eg, sg, i_all, j_all, out, 4 * EH,
                                              out + 2 * EH, 4 * EH, 2 * EH / 4,
                                              N_EALL);
        leaky_slab<<<cdiv((long long)N_NODES * 2 * EH, 256), 256, 0, stream>>>(
            out, N_NODES, 4 * EH, 2 * EH, 2 * EH);
    }
}